// AdjacencyErrorAwareLoss_816043786443
// MI455X (gfx1250) — compile-verified
//
#include <hip/hip_runtime.h>
#include <stdint.h>

// Problem constants from the reference: B=64, NL=128, NP=128, E=4096
#define B_   64
#define NL_  128
#define NP_  128
#define E_   4096

typedef __attribute__((ext_vector_type(2))) float v2f;
typedef __attribute__((ext_vector_type(8))) float v8f;

// ---- CDNA5 async global->LDS DMA (ASYNCcnt path, ISA ch.10 / 15.18) ----
__device__ __forceinline__ void async_copy_b128(uint32_t lds_byte_off, const float* gsrc) {
  asm volatile("global_load_async_to_lds_b128 %0, %1, off"
               :: "v"(lds_byte_off), "v"((const void*)gsrc)
               : "memory");
}
__device__ __forceinline__ void wait_asynccnt0() {
  asm volatile("s_wait_asynccnt 0" ::: "memory");
}

// Fused per-batch kernel:
//  phase 0: DMA P[b] (64KB) into LDS; build A_fid^T = ((d_hw==1)*max(1-d_error,0))^T in LDS
//           (transpose taken on the global-read side -> conflict-free LDS writes)
//  phase 1: Q = P[b] @ A_fid via V_WMMA_F32_16X16X4_F32 (full f32 precision);
//           B fragments are aligned ds_load_b64 from A_fid^T, all-immediate offsets
//  phase 2: per-edge score = Q[i,:]·P[j,:] from LDS; weighted accumulate; sample_adj[b]
__global__ __launch_bounds__(256) void adj_loss_fused_kernel(
    const float* __restrict__ P,        // [B, NL, NP]
    const float* __restrict__ d_error,  // [NP, NP]
    const float* __restrict__ ew,       // [B, E]
    const int*   __restrict__ d_hw,     // [NP, NP]
    const long long* __restrict__ pairs,// [B, E, 2]
    float* __restrict__ perBatch)       // [B] sample_adj
{
  extern __shared__ float lds[];
  float* sAT  = lds;           // A_fid^T [q][p]   (16384 f)
  float* sP   = lds + 16384;   // P[b]    [l][p]   (16384 f)
  float* sQ   = lds + 32768;   // Q       [l][q]   (16384 f)
  float* sRed = lds + 49152;   // 16 floats reduction scratch

  const int b    = blockIdx.x;
  const int t    = threadIdx.x;
  const int wave = t >> 5;      // 8 waves (wave32)
  const int lane = t & 31;
  const int m    = lane & 15;
  const int hi   = lane >> 4;   // 0 or 1

  // ---------------- phase 0a: stage P[b] into LDS via async DMA ----------------
  const float* Pg = P + (size_t)b * NL_ * NP_;
  const uint32_t sP_base = (uint32_t)(uintptr_t)sP;  // low 32 bits of flat = LDS byte offset
  #pragma unroll
  for (int i = 0; i < 16; ++i) {
    const int e4 = (i * 256 + t) * 4;                 // float index, lanes contiguous (16B/lane)
    async_copy_b128(sP_base + (uint32_t)e4 * 4u, Pg + e4);
  }

  // ---------------- phase 0b: A_fid^T while DMA is in flight ----------------
  // sAT[q*128+p] = A_fid[p][q]; write index contiguous per thread (no LDS bank conflicts),
  // transpose absorbed by the (L2-resident) global read of d_hw/d_error.
  #pragma unroll 4
  for (int k = 0; k < 64; ++k) {
    const int idx = t + k * 256;       // = q*128 + p
    const int q   = idx >> 7;
    const int p   = idx & 127;
    const int g   = p * NP_ + q;
    sAT[idx] = (d_hw[g] == 1) ? fmaxf(1.0f - d_error[g], 0.0f) : 0.0f;
  }

  wait_asynccnt0();
  __syncthreads();

  // ---------------- phase 1: Q = sP @ A_fid with f32 WMMA ----------------
  // wave owns rows [strip, strip+16); 8 column tiles of 16 -> 8 accumulators
  const int strip = wave * 16;
  v8f acc[8] = {};

  // A fragment (32-bit A 16x4, ISA 7.12.2): lane m holds row strip+m,
  //   v0 -> K = k0 + 2*hi, v1 -> K = k0 + 2*hi + 1   (one aligned ds_load_b64)
  // B fragment (4x16): VGPR r -> K = k0 + r + 2*hi, N = n*16 + m; from sAT this is
  //   sAT[(n*16+m)*128 + k0 + 2*hi] .. +1  (one aligned ds_load_b64, offset n*8192+kk*16+hi*8
  //   fits the 16-bit DS immediate -> single address register for the whole GEMM)
  const float* aRow = sP  + (strip + m) * NP_ + 2 * hi;   // + kk*4 via immediate
  const float* bCol = sAT + m * NP_ + 2 * hi;             // + n*2048 + kk*4 via immediate

  #pragma unroll
  for (int kk = 0; kk < 32; ++kk) {
    const int k0 = kk * 4;
    v2f afrag = *(const v2f*)(aRow + k0);
    #pragma unroll
    for (int n = 0; n < 8; ++n) {
      v2f bfrag = *(const v2f*)(bCol + n * 16 * NP_ + k0);
      acc[n] = __builtin_amdgcn_wmma_f32_16x16x4_f32(
          false, afrag, false, bfrag, (short)0, acc[n], false, false);
    }
  }

  // C/D layout (8 VGPRs): row = strip + r + 8*hi, col = n*16 + m
  #pragma unroll
  for (int n = 0; n < 8; ++n) {
    #pragma unroll
    for (int r = 0; r < 8; ++r) {
      sQ[(strip + r + 8 * hi) * NP_ + n * 16 + m] = acc[n][r];
    }
  }
  __syncthreads();

  // ---------------- phase 2: weighted edge scores ----------------
  // sum_e w_e * (Q[i]·P[j]) = sum_lane sum_e w_e * partial_lane  -> one reduction at end
  const long long* pb  = pairs + (size_t)b * E_ * 2;
  const float*     ewb = ew    + (size_t)b * E_;
  const int c0 = lane * 4;

  float sAcc = 0.0f;
  const int eBeg = wave * (E_ / 8), eEnd = eBeg + (E_ / 8);
  for (int e = eBeg; e < eEnd; ++e) {
    const int i = (int)pb[2 * e];        // wave-uniform -> scalarizable
    const int j = (int)pb[2 * e + 1];
    const float w = ewb[e];
    const float* qi = sQ + i * NP_ + c0; // ds_load_b128 per lane
    const float* pj = sP + j * NP_ + c0;
    const float part = qi[0] * pj[0] + qi[1] * pj[1] + qi[2] * pj[2] + qi[3] * pj[3];
    sAcc += w * part;
  }

  // per-thread partial of w_sum
  float wAcc = 0.0f;
  #pragma unroll
  for (int e = t; e < E_; e += 256) wAcc += ewb[e];

  // wave32 reductions
  #pragma unroll
  for (int off = 16; off >= 1; off >>= 1) {
    sAcc += __shfl_xor(sAcc, off, 32);
    wAcc += __shfl_xor(wAcc, off, 32);
  }
  if (lane == 0) { sRed[wave] = sAcc; sRed[8 + wave] = wAcc; }
  __syncthreads();

  if (t == 0) {
    float S = 0.0f, W = 0.0f;
    #pragma unroll
    for (int w = 0; w < 8; ++w) { S += sRed[w]; W += sRed[8 + w]; }
    perBatch[b] = S / fmaxf(W, 1e-8f);
  }
}

// loss = -mean_b(sample_adj[b]); single wave, deterministic (no float atomics)
__global__ __launch_bounds__(32) void adj_loss_finalize_kernel(
    const float* __restrict__ perBatch, float* __restrict__ out)
{
  const int l = threadIdx.x;
  float v = perBatch[l] + perBatch[l + 32];
  #pragma unroll
  for (int off = 16; off >= 1; off >>= 1) v += __shfl_xor(v, off, 32);
  if (l == 0) out[0] = -v / (float)B_;
}

extern "C" void kernel_launch(void* const* d_in, const int* in_sizes, int n_in,
                              void* d_out, int out_size, void* d_ws, size_t ws_size,
                              hipStream_t stream) {
  const float*     P       = (const float*)d_in[0];
  const float*     d_error = (const float*)d_in[1];
  const float*     ew      = (const float*)d_in[2];
  const int*       d_hw    = (const int*)d_in[3];
  const long long* pairs   = (const long long*)d_in[4];
  float* perBatch = (float*)d_ws;          // 64 floats of scratch
  float* out      = (float*)d_out;

  const size_t ldsBytes = (size_t)(49152 + 16) * sizeof(float);  // ~192KB of the 320KB/WGP
  (void)hipFuncSetAttribute((const void*)adj_loss_fused_kernel,
                            hipFuncAttributeMaxDynamicSharedMemorySize,
                            (int)ldsBytes);

  adj_loss_fused_kernel<<<B_, 256, ldsBytes, stream>>>(P, d_error, ew, d_hw, pairs, perBatch);
  adj_loss_finalize_kernel<<<1, 32, 0, stream>>>(perBatch, out);
}